// ShapeletEncoder_14577119003491
// MI455X (gfx1250) — compile-verified
//
#include <hip/hip_runtime.h>

typedef float v8f __attribute__((ext_vector_type(8)));
typedef float v2f __attribute__((ext_vector_type(2)));

#define THREADS 256
#define ROWS    16
#define NWAVE   8
#define CH_PER_WAVE 4

static __device__ __forceinline__ v8f wmma_f32(v2f a, v2f b, v8f c) {
    // D = A(16x4,f32) * B(4x16,f32) + C(16x16,f32), full fp32 precision
    return __builtin_amdgcn_wmma_f32_16x16x4_f32(false, a, false, b, (short)0, c, false, false);
}

static __device__ __forceinline__ void wave_fence() {
#if defined(__has_builtin)
#if __has_builtin(__builtin_amdgcn_wave_barrier)
    __builtin_amdgcn_wave_barrier();
#endif
#endif
}

__global__ __launch_bounds__(THREADS)
void shapelet_encoder_kernel(const float* __restrict__ x,
                             const float* __restrict__ W0, const float* __restrict__ b0,
                             const float* __restrict__ W1, const float* __restrict__ b1,
                             const float* __restrict__ W2, const float* __restrict__ b2,
                             const float* __restrict__ Wf, const float* __restrict__ bf,
                             float* __restrict__ out)
{
    __shared__ float xbuf[ROWS * 512];          // 32 KB: 16 rows x (L=16, C=32)
    __shared__ float mid [ROWS * 32];           //  2 KB: midpoints per (row, c)
    __shared__ float hA  [NWAVE][ROWS * 32];    // 16 KB: per-wave activations (ping)
    __shared__ float hB  [NWAVE][ROWS * 32];    // 16 KB: per-wave activations (pong)
    __shared__ float outbuf[ROWS * 320];        // 20 KB: staged contiguous output tile

    const int tid  = threadIdx.x;
    const int wave = tid >> 5;
    const int lid  = tid & 31;
    const int half = lid >> 4;     // 0: lanes 0-15, 1: lanes 16-31
    const int l16  = lid & 15;

    // ---- stage contiguous 32 KB x tile (coalesced float4) ----
    {
        const float4* src = (const float4*)(x) + (size_t)blockIdx.x * (ROWS * 512 / 4);
        float4* dst = (float4*)xbuf;
        #pragma unroll
        for (int i = 0; i < 8; ++i)
            dst[tid + i * THREADS] = src[tid + i * THREADS];
    }
    __syncthreads();

    // ---- per-(row,channel) min-max midpoint: 512 tasks over 256 threads ----
    #pragma unroll
    for (int t = 0; t < 2; ++t) {
        const int task = tid + t * THREADS;
        const int row = task >> 5, c = task & 31;
        float mn = xbuf[row * 512 + c], mx = mn;
        #pragma unroll
        for (int l = 1; l < 16; ++l) {
            const float v = xbuf[row * 512 + l * 32 + c];
            mn = fminf(mn, v); mx = fmaxf(mx, v);
        }
        mid[row * 32 + c] = 0.5f * (mn + mx);
    }
    __syncthreads();

    float* HA = hA[wave];
    float* HB = hB[wave];

    for (int ci = 0; ci < CH_PER_WAVE; ++ci) {
        const int c = wave * CH_PER_WAVE + ci;

        // ======== layer 0: norm[16x16] @ W0[c](16x32) + b0, ReLU -> HA[16x32] ========
        {
            v8f acc0, acc1;
            const float bia0 = b0[c * 32 + l16];
            const float bia1 = b0[c * 32 + 16 + l16];
            #pragma unroll
            for (int r = 0; r < 8; ++r) { acc0[r] = bia0; acc1[r] = bia1; }
            const float m = mid[l16 * 32 + c];
            const float* w = W0 + c * 512;                 // (K=16) x (N=32)
            #pragma unroll
            for (int k = 0; k < 4; ++k) {
                const int k0 = 4 * k + 2 * half;
                v2f a, bf0, bf1;
                a.x = xbuf[l16 * 512 + k0 * 32 + c] - m;
                a.y = xbuf[l16 * 512 + (k0 + 1) * 32 + c] - m;
                bf0.x = w[k0 * 32 + l16];       bf0.y = w[(k0 + 1) * 32 + l16];
                bf1.x = w[k0 * 32 + 16 + l16];  bf1.y = w[(k0 + 1) * 32 + 16 + l16];
                acc0 = wmma_f32(a, bf0, acc0);
                acc1 = wmma_f32(a, bf1, acc1);
            }
            #pragma unroll
            for (int r = 0; r < 8; ++r) {
                const int row = r + 8 * half;
                HA[row * 32 + l16]      = fmaxf(acc0[r], 0.0f);
                HA[row * 32 + 16 + l16] = fmaxf(acc1[r], 0.0f);
            }
        }
        wave_fence();

        // ======== layer 1: HA[16x32] @ W1[c](32x32) + b1, ReLU -> HB[16x32] ========
        {
            v8f acc0, acc1;
            const float bia0 = b1[c * 32 + l16];
            const float bia1 = b1[c * 32 + 16 + l16];
            #pragma unroll
            for (int r = 0; r < 8; ++r) { acc0[r] = bia0; acc1[r] = bia1; }
            const float* w = W1 + c * 1024;                // (K=32) x (N=32)
            #pragma unroll
            for (int k = 0; k < 8; ++k) {
                const int k0 = 4 * k + 2 * half;
                v2f a, bf0, bf1;
                a.x = HA[l16 * 32 + k0];
                a.y = HA[l16 * 32 + k0 + 1];
                bf0.x = w[k0 * 32 + l16];       bf0.y = w[(k0 + 1) * 32 + l16];
                bf1.x = w[k0 * 32 + 16 + l16];  bf1.y = w[(k0 + 1) * 32 + 16 + l16];
                acc0 = wmma_f32(a, bf0, acc0);
                acc1 = wmma_f32(a, bf1, acc1);
            }
            #pragma unroll
            for (int r = 0; r < 8; ++r) {
                const int row = r + 8 * half;
                HB[row * 32 + l16]      = fmaxf(acc0[r], 0.0f);
                HB[row * 32 + 16 + l16] = fmaxf(acc1[r], 0.0f);
            }
        }
        wave_fence();

        // ======== layer 2: HB[16x32] @ W2[c](32x16) + b2, ReLU -> HA[16x16] ========
        {
            v8f acc;
            const float bia = b2[c * 16 + l16];
            #pragma unroll
            for (int r = 0; r < 8; ++r) acc[r] = bia;
            const float* w = W2 + c * 512;                 // (K=32) x (N=16)
            #pragma unroll
            for (int k = 0; k < 8; ++k) {
                const int k0 = 4 * k + 2 * half;
                v2f a, bfr;
                a.x = HB[l16 * 32 + k0];
                a.y = HB[l16 * 32 + k0 + 1];
                bfr.x = w[k0 * 16 + l16];
                bfr.y = w[(k0 + 1) * 16 + l16];
                acc = wmma_f32(a, bfr, acc);
            }
            #pragma unroll
            for (int r = 0; r < 8; ++r) {
                const int row = r + 8 * half;
                HA[row * 16 + l16] = fmaxf(acc[r], 0.0f);   // 16x16, stride 16
            }
        }
        wave_fence();

        // ======== layer 3: HA[16x16] @ Wf[c](16x10 pad 16) + bf -> logits HB[16x16] ========
        {
            v8f acc;
            const bool valid = (l16 < 10);
            const float bia = valid ? bf[c * 10 + l16] : 0.0f;
            #pragma unroll
            for (int r = 0; r < 8; ++r) acc[r] = bia;
            const float* w = Wf + c * 160;                 // (K=16) x (N=10)
            #pragma unroll
            for (int k = 0; k < 4; ++k) {
                const int k0 = 4 * k + 2 * half;
                v2f a, bfr;
                a.x = HA[l16 * 16 + k0];
                a.y = HA[l16 * 16 + k0 + 1];
                bfr.x = valid ? w[k0 * 10 + l16] : 0.0f;
                bfr.y = valid ? w[(k0 + 1) * 10 + l16] : 0.0f;
                acc = wmma_f32(a, bfr, acc);
            }
            #pragma unroll
            for (int r = 0; r < 8; ++r) {
                const int row = r + 8 * half;
                HB[row * 16 + l16] = acc[r];               // cols >=10 are garbage, unused
            }
        }
        wave_fence();

        // ======== softmax over 10 classes: lanes 0-15 each take one row ========
        if (lid < 16) {
            const int row = lid;
            float v[10], mx = HB[row * 16 + 0];
            #pragma unroll
            for (int j = 0; j < 10; ++j) v[j] = HB[row * 16 + j];
            #pragma unroll
            for (int j = 1; j < 10; ++j) mx = fmaxf(mx, v[j]);
            float s = 0.0f;
            #pragma unroll
            for (int j = 0; j < 10; ++j) { v[j] = __expf(v[j] - mx); s += v[j]; }
            const float inv = 1.0f / s;
            #pragma unroll
            for (int j = 0; j < 10; ++j)
                outbuf[row * 320 + c * 10 + j] = v[j] * inv;
        }
        wave_fence();
    }

    __syncthreads();

    // ---- coalesced contiguous store of 20 KB output tile ----
    {
        float4* dst = (float4*)(out) + (size_t)blockIdx.x * (ROWS * 320 / 4);
        const float4* src = (const float4*)outbuf;
        #pragma unroll
        for (int i = 0; i < 5; ++i)
            dst[tid + i * THREADS] = src[tid + i * THREADS];
    }
}

extern "C" void kernel_launch(void* const* d_in, const int* in_sizes, int n_in,
                              void* d_out, int out_size, void* d_ws, size_t ws_size,
                              hipStream_t stream) {
    const float* x  = (const float*)d_in[0];
    const float* W0 = (const float*)d_in[1];
    const float* b0 = (const float*)d_in[2];
    const float* W1 = (const float*)d_in[3];
    const float* b1 = (const float*)d_in[4];
    const float* W2 = (const float*)d_in[5];
    const float* b2 = (const float*)d_in[6];
    const float* Wf = (const float*)d_in[7];
    const float* bfv = (const float*)d_in[8];
    float* out = (float*)d_out;

    // rows = B*CELLS; each row holds L*C = 512 floats of x
    const int rows = in_sizes[0] / 512;      // 131072
    const int blocks = rows / ROWS;          // 8192
    shapelet_encoder_kernel<<<blocks, THREADS, 0, stream>>>(
        x, W0, b0, W1, b1, W2, b2, Wf, bfv, out);
}